// GCN_83133386981892
// MI455X (gfx1250) — compile-verified
//
#include <hip/hip_runtime.h>
#include <hip/hip_bf16.h>

#define NN 100000
#define NE 1600000
#define CH 128
#define CLS 64
#define NG 256

typedef __attribute__((ext_vector_type(2))) float v2f;
typedef __attribute__((ext_vector_type(8))) float v8f;

// ---------------------------------------------------------------------------
// H[M x 128] = X[M x 128] @ W[128 x 128]  via V_WMMA_F32_16X16X4_F32
// grid.x = M/16 (m-tile), blockDim = (32,8): threadIdx.y selects the n-tile.
// Layouts per CDNA5 ISA 7.12.2:
//   A 16x4 f32 : lane&15 = M row; v0 = A[M][2*(lane>>4)], v1 = A[M][2*(lane>>4)+1]
//   B 4x16 f32 : lane&15 = N col; v0 = B[2*(lane>>4)][N], v1 = B[2*(lane>>4)+1][N]
//   C/D 16x16  : acc[i] = D[8*(lane>>4) + i][lane&15]
// ---------------------------------------------------------------------------
__global__ __launch_bounds__(256)
void gemm128_wmma(const float* __restrict__ X, const float* __restrict__ W,
                  float* __restrict__ H) {
  const int m0   = blockIdx.x * 16;
  const int n0   = threadIdx.y * 16;
  const int lane = threadIdx.x;
  const int idx  = lane & 15;      // M row for A, N col for B/D
  const int kh   = lane >> 4;      // K half-select

  v8f acc = {0.f,0.f,0.f,0.f,0.f,0.f,0.f,0.f};
  const float* xrow = X + (size_t)(m0 + idx) * CH;

#pragma unroll
  for (int k = 0; k < CH; k += 4) {
    const int ka = k + 2 * kh;
    v2f a, b;
    a.x = xrow[ka];
    a.y = xrow[ka + 1];
    b.x = W[(size_t)ka * CH + n0 + idx];
    b.y = W[(size_t)(ka + 1) * CH + n0 + idx];
    acc = __builtin_amdgcn_wmma_f32_16x16x4_f32(false, a, false, b,
                                                (short)0, acc, false, false);
  }

  float* hcol = H + (size_t)(m0 + 8 * kh) * CH + n0 + idx;
#pragma unroll
  for (int i = 0; i < 8; ++i) hcol[(size_t)i * CH] = acc[i];
}

// ---------------------------------------------------------------------------
// deg[dst] += 1 over edges (deg buffer pre-zeroed)
// ---------------------------------------------------------------------------
__global__ void deg_count(const int* __restrict__ dst, float* __restrict__ deg) {
  int e = blockIdx.x * blockDim.x + threadIdx.x;
  if (e < NE) atomicAdd(&deg[dst[e]], 1.0f);
}

// deg -> dinv = rsqrt(deg + 1) in place
__global__ void deg_to_dinv(float* __restrict__ d) {
  int i = blockIdx.x * blockDim.x + threadIdx.x;
  if (i < NN) d[i] = rsqrtf(d[i] + 1.0f);
}

// ---------------------------------------------------------------------------
// out = h * dinv[node]^2 + bias[ch]   (self-loop term + bias), float4 lanes
// ---------------------------------------------------------------------------
__global__ void self_bias(const float* __restrict__ h, const float* __restrict__ dinv,
                          const float* __restrict__ b, float* __restrict__ out) {
  int i = blockIdx.x * blockDim.x + threadIdx.x;       // float4 index
  if (i >= NN * (CH / 4)) return;
  float di = dinv[i >> 5];
  float d2 = di * di;
  float4 hv = ((const float4*)h)[i];
  float4 bv = ((const float4*)b)[i & 31];
  float4 o;
  o.x = hv.x * d2 + bv.x;  o.y = hv.y * d2 + bv.y;
  o.z = hv.z * d2 + bv.z;  o.w = hv.w * d2 + bv.w;
  ((float4*)out)[i] = o;
}

// ---------------------------------------------------------------------------
// One wave per edge: gather h[src] (float4/lane), scale, scatter-add to out[dst]
// ---------------------------------------------------------------------------
__global__ __launch_bounds__(256)
void edge_scatter(const float* __restrict__ h, float* __restrict__ out,
                  const int* __restrict__ src, const int* __restrict__ dst,
                  const float* __restrict__ dinv) {
  int e = blockIdx.x * blockDim.y + threadIdx.y;
  if (e >= NE) return;
  int lane = threadIdx.x;
  int s = src[e], d = dst[e];
  float norm = dinv[s] * dinv[d];
  float4 v = ((const float4*)(h + (size_t)s * CH))[lane];
  float* od = out + (size_t)d * CH + lane * 4;
  atomicAdd(od + 0, v.x * norm);
  atomicAdd(od + 1, v.y * norm);
  atomicAdd(od + 2, v.z * norm);
  atomicAdd(od + 3, v.w * norm);
}

// in-place ReLU, float4
__global__ void relu_ip(float* __restrict__ p) {
  int i = blockIdx.x * blockDim.x + threadIdx.x;
  if (i >= NN * (CH / 4)) return;
  float4 v = ((float4*)p)[i];
  v.x = fmaxf(v.x, 0.f); v.y = fmaxf(v.y, 0.f);
  v.z = fmaxf(v.z, 0.f); v.w = fmaxf(v.w, 0.f);
  ((float4*)p)[i] = v;
}

// ---------------------------------------------------------------------------
// Pool: one wave per node. ReLU applied on read (layer-2 ReLU fused here).
// Max via u32 atomicMax on bit pattern (valid: post-ReLU values >= 0).
// ---------------------------------------------------------------------------
__global__ __launch_bounds__(256)
void pool_kernel(const float* __restrict__ h, const int* __restrict__ batch,
                 float* __restrict__ padd, unsigned* __restrict__ pmax,
                 float* __restrict__ pcnt) {
  int n = blockIdx.x * blockDim.y + threadIdx.y;
  if (n >= NN) return;
  int lane = threadIdx.x;
  int g = batch[n];
  float4 v = ((const float4*)(h + (size_t)n * CH))[lane];
  v.x = fmaxf(v.x, 0.f); v.y = fmaxf(v.y, 0.f);
  v.z = fmaxf(v.z, 0.f); v.w = fmaxf(v.w, 0.f);
  float*    pa = padd + (size_t)g * CH + lane * 4;
  unsigned* pm = pmax + (size_t)g * CH + lane * 4;
  atomicAdd(pa + 0, v.x); atomicAdd(pa + 1, v.y);
  atomicAdd(pa + 2, v.z); atomicAdd(pa + 3, v.w);
  atomicMax(pm + 0, __float_as_uint(v.x));
  atomicMax(pm + 1, __float_as_uint(v.y));
  atomicMax(pm + 2, __float_as_uint(v.z));
  atomicMax(pm + 3, __float_as_uint(v.w));
  if (lane == 0) atomicAdd(&pcnt[g], 1.0f);
}

// hid[g][j] = relu( concat(mean,max,add)[g] . Wl[:,j] + bl[j] )
__global__ void head1(const float* __restrict__ padd, const float* __restrict__ pmax,
                      const float* __restrict__ pcnt, const float* __restrict__ Wl,
                      const float* __restrict__ bl, float* __restrict__ hid) {
  int t = blockIdx.x * blockDim.x + threadIdx.x;
  if (t >= NG * CLS) return;
  int g = t / CLS, j = t % CLS;
  float inv = 1.0f / fmaxf(pcnt[g], 1.0f);
  const float* ad = padd + (size_t)g * CH;
  const float* mx = pmax + (size_t)g * CH;
  float acc = bl[j];
  for (int k = 0; k < CH; ++k) {
    float a = ad[k];
    acc += (a * inv) * Wl[(size_t)k * CLS + j];          // mean
    acc += mx[k]     * Wl[(size_t)(CH + k) * CLS + j];   // max
    acc += a         * Wl[(size_t)(2 * CH + k) * CLS + j]; // add
  }
  hid[t] = fmaxf(acc, 0.f);
}

// out[g] = sigmoid( hid[g] . Wl2 + bl2 )
__global__ void head2(const float* __restrict__ hid, const float* __restrict__ Wl2,
                      const float* __restrict__ bl2, float* __restrict__ out) {
  int g = blockIdx.x * blockDim.x + threadIdx.x;
  if (g >= NG) return;
  float acc = bl2[0];
  for (int j = 0; j < CLS; ++j) acc += hid[(size_t)g * CLS + j] * Wl2[j];
  out[g] = 1.0f / (1.0f + expf(-acc));
}

// ---------------------------------------------------------------------------
extern "C" void kernel_launch(void* const* d_in, const int* in_sizes, int n_in,
                              void* d_out, int out_size, void* d_ws, size_t ws_size,
                              hipStream_t stream) {
  const float* x    = (const float*)d_in[0];
  const int*   ei   = (const int*)d_in[1];
  const int*   src  = ei;
  const int*   dst  = ei + NE;
  const int*   batch= (const int*)d_in[3];
  const float* W1   = (const float*)d_in[4];
  const float* b1   = (const float*)d_in[5];
  const float* W2   = (const float*)d_in[6];
  const float* b2   = (const float*)d_in[7];
  const float* Wl   = (const float*)d_in[8];
  const float* bl   = (const float*)d_in[9];
  const float* Wl2  = (const float*)d_in[10];
  const float* bl2  = (const float*)d_in[11];

  char* ws = (char*)d_ws;
  const size_t HB = (size_t)NN * CH * sizeof(float);      // 51.2 MB
  float*    hA   = (float*)(ws);                          // GEMM output
  float*    hB   = (float*)(ws + HB);                     // aggregated output
  float*    dinv = (float*)(ws + 2 * HB);                 // deg -> dinv (400 KB)
  char*     p    = ws + 2 * HB + ((NN * sizeof(float) + 255) & ~(size_t)255);
  float*    padd = (float*)p;              p += (size_t)NG * CH * sizeof(float);
  unsigned* pmax = (unsigned*)p;           p += (size_t)NG * CH * sizeof(unsigned);
  float*    pcnt = (float*)p;              p += (size_t)NG * sizeof(float);
  float*    hid  = (float*)p;

  const int THR = 256;
  const dim3 wave8(32, 8);

  // degrees -> dinv
  hipMemsetAsync(dinv, 0, NN * sizeof(float), stream);
  deg_count  <<<(NE + THR - 1) / THR, THR, 0, stream>>>(dst, dinv);
  deg_to_dinv<<<(NN + THR - 1) / THR, THR, 0, stream>>>(dinv);

  const int elem4 = NN * (CH / 4);

  // ---- layer 1 ----
  gemm128_wmma<<<NN / 16, wave8, 0, stream>>>(x, W1, hA);
  self_bias   <<<(elem4 + THR - 1) / THR, THR, 0, stream>>>(hA, dinv, b1, hB);
  edge_scatter<<<(NE + 7) / 8, wave8, 0, stream>>>(hA, hB, src, dst, dinv);
  relu_ip     <<<(elem4 + THR - 1) / THR, THR, 0, stream>>>(hB);

  // ---- layer 2 (ReLU fused into pooling read) ----
  gemm128_wmma<<<NN / 16, wave8, 0, stream>>>(hB, W2, hA);
  self_bias   <<<(elem4 + THR - 1) / THR, THR, 0, stream>>>(hA, dinv, b2, hB);
  edge_scatter<<<(NE + 7) / 8, wave8, 0, stream>>>(hA, hB, src, dst, dinv);

  // ---- pooling + head ----
  hipMemsetAsync(padd, 0,
                 (size_t)NG * CH * sizeof(float) * 2 + NG * sizeof(float), stream);
  pool_kernel<<<(NN + 7) / 8, wave8, 0, stream>>>(hB, batch, padd, pmax, pcnt);
  head1<<<(NG * CLS + THR - 1) / THR, THR, 0, stream>>>(padd, (const float*)pmax,
                                                        pcnt, Wl, bl, hid);
  head2<<<(NG + THR - 1) / THR, THR, 0, stream>>>(hid, Wl2, bl2, (float*)d_out);
}